// MagnitudeMaxPool_27625229648238
// MI455X (gfx1250) — compile-verified
//
#include <hip/hip_runtime.h>

// MagnitudeMaxPool (complex 2x2/stride-2 max-|x| pool), MI455X / gfx1250.
//
// Roofline: 640 MiB streamed once (512 in + 128 out) -> ~27.5 us at 23.3 TB/s.
// Pure HBM-bound; WMMA is inapplicable (argmax-select, no matmul structure) and
// LDS/TDM staging adds traffic with zero reuse. Optimal path = wave32-coalesced
// global_load_b128 / global_store_b128 with non-temporal (TH_NT) cache policy
// so the one-pass stream does not thrash the 192 MB L2, plus branchless
// v_cmp/v_cndmask selection.

typedef float __attribute__((ext_vector_type(4))) v4f;

namespace {
constexpr int      B   = 8;
constexpr int      Hh  = 512;
constexpr int      Ww  = 512;
constexpr int      C   = 32;
constexpr int      Ho  = Hh / 2;                       // 256
constexpr int      Wo  = Ww / 2;                       // 256
constexpr int      CQ  = C / 4;                        // 8 channel-quads / window
constexpr unsigned TOTAL = (unsigned)B * Ho * Wo * CQ; // 4,194,304 threads
constexpr size_t   ROW = (size_t)Ww * C;               // 16384 floats per input row
constexpr size_t   IMG = (size_t)Hh * Ww * C;          // 8,388,608 floats per image
} // namespace

__global__ __launch_bounds__(256) void MagnitudeMaxPool_kernel(
    const float* __restrict__ xr, const float* __restrict__ xi,
    float* __restrict__ out) {
  unsigned t = blockIdx.x * blockDim.x + threadIdx.x;
  if (t >= TOTAL) return;

  // lane = low bits => channel-quad fastest: coalesced b128 across the wave.
  unsigned cq = t & (CQ - 1);          // 0..7
  unsigned wo = (t >> 3) & (Wo - 1);   // 0..255
  unsigned ho = (t >> 11) & (Ho - 1);  // 0..255
  unsigned b  = t >> 19;               // 0..7

  size_t base = (size_t)b * IMG + (size_t)(2u * ho) * ROW +
                (size_t)(2u * wo) * C + 4u * cq;
  const v4f* pr = (const v4f*)(xr + base);
  const v4f* pi = (const v4f*)(xi + base);

  // v4f-unit offsets of the 2x2 window positions (NHWC, C=32):
  //   (h, w)    -> 0
  //   (h, w+1)  -> C/4       = 8
  //   (h+1, w)  -> ROW/4     = 4096
  //   (h+1, w+1)-> ROW/4 + 8 = 4104
  // Position order (0,0),(0,1),(1,0),(1,1) matches the reference's P*P axis,
  // so strict '>' below reproduces jnp.argmax first-max tie-breaking.
  v4f r0 = __builtin_nontemporal_load(pr);
  v4f r1 = __builtin_nontemporal_load(pr + 8);
  v4f r2 = __builtin_nontemporal_load(pr + 4096);
  v4f r3 = __builtin_nontemporal_load(pr + 4104);
  v4f i0 = __builtin_nontemporal_load(pi);
  v4f i1 = __builtin_nontemporal_load(pi + 8);
  v4f i2 = __builtin_nontemporal_load(pi + 4096);
  v4f i3 = __builtin_nontemporal_load(pi + 4104);

  v4f m0 = r0 * r0 + i0 * i0;   // v_mul + v_fma per element
  v4f m1 = r1 * r1 + i1 * i1;
  v4f m2 = r2 * r2 + i2 * i2;
  v4f m3 = r3 * r3 + i3 * i3;

  v4f lo, hi;  // interleaved (r,i): lo = {r_c0,i_c0,r_c1,i_c1}, hi = c2,c3
#pragma unroll
  for (int j = 0; j < 4; ++j) {
    float br = r0[j], bi = i0[j], bm = m0[j];
    if (m1[j] > bm) { br = r1[j]; bi = i1[j]; bm = m1[j]; }  // v_cndmask x3
    if (m2[j] > bm) { br = r2[j]; bi = i2[j]; bm = m2[j]; }
    if (m3[j] > bm) { br = r3[j]; bi = i3[j]; }              // bm dead after
    if (j < 2) {
      lo[2 * j]     = br;
      lo[2 * j + 1] = bi;
    } else {
      hi[2 * (j - 2)]     = br;
      hi[2 * (j - 2) + 1] = bi;
    }
  }

  // out[b][ho][wo][c][2]; 4 channels x (r,i) = 32 contiguous bytes.
  size_t ob = ((((size_t)b * Ho + ho) * Wo + wo) * C + 4u * cq) * 2u;
  v4f* po = (v4f*)(out + ob);
  __builtin_nontemporal_store(lo, po);
  __builtin_nontemporal_store(hi, po + 1);
}

extern "C" void kernel_launch(void* const* d_in, const int* in_sizes, int n_in,
                              void* d_out, int out_size, void* d_ws,
                              size_t ws_size, hipStream_t stream) {
  const float* xr = (const float*)d_in[0];
  const float* xi = (const float*)d_in[1];
  float* out = (float*)d_out;

  dim3 block(256);                 // 8 wave32 waves per block
  dim3 grid(TOTAL / 256);          // 16384 blocks, exact cover
  hipLaunchKernelGGL(MagnitudeMaxPool_kernel, grid, block, 0, stream, xr, xi,
                     out);
}